// GIN_18940805775883
// MI455X (gfx1250) — compile-verified
//
#include <hip/hip_runtime.h>

#define COUT 128

typedef float v2f __attribute__((ext_vector_type(2)));
typedef float v8f __attribute__((ext_vector_type(8)));

// ---------------------------------------------------------------- copy (init agg = h)
__global__ void copy_f4(float* __restrict__ dst, const float* __restrict__ src, int n4) {
  int i = blockIdx.x * blockDim.x + threadIdx.x;
  if (i < n4) reinterpret_cast<float4*>(dst)[i] = reinterpret_cast<const float4*>(src)[i];
}

// ------------------------------------------------- edge scatter: agg[dst] += h[src]
// one thread per (edge, 4-channel group); 4 scalar f32 atomics -> global_atomic_add_f32
__global__ void scatter_add(float* __restrict__ agg, const float* __restrict__ h,
                            const int* __restrict__ src, const int* __restrict__ dst,
                            int total, int gshift, int C) {
  int idx = blockIdx.x * blockDim.x + threadIdx.x;
  if (idx >= total) return;
  int e = idx >> gshift;
  int g = idx & ((1 << gshift) - 1);
  int s = src[e], d = dst[e];
  float4 v = reinterpret_cast<const float4*>(h + (size_t)s * C)[g];
  float* p = agg + (size_t)d * C + (g << 2);
  atomicAdd(p + 0, v.x);
  atomicAdd(p + 1, v.y);
  atomicAdd(p + 2, v.z);
  atomicAdd(p + 3, v.w);
}

// ---------------------------------------------------------------- WMMA f32 GEMM
// out[N, 128] = act(A[N, K] @ W[K, 128] + bias), K in {64, 128}
// Block: 256 threads = 8 waves; wave w computes rows [ (blk*8+w)*16 , +16 ) x all 128 cols.
// W staged in LDS ([k][col], row-major, K*128 floats <= 64 KB), shared by all waves.
__global__ __launch_bounds__(256)
void gemm_bias_act(const float* __restrict__ A, const float* __restrict__ W,
                   const float* __restrict__ bias, float* __restrict__ out,
                   int N, int K, int do_relu) {
  extern __shared__ float lds[];
  for (int i = threadIdx.x; i < K * COUT; i += blockDim.x) lds[i] = W[i];
  __syncthreads();

  const int wave = threadIdx.x >> 5;
  const int lane = threadIdx.x & 31;
  const int half = lane >> 4;   // 0: K pair {k0,k0+1}; 1: {k0+2,k0+3}
  const int l15  = lane & 15;
  const int row0 = (blockIdx.x * 8 + wave) * 16;
  if (row0 >= N) return;

  // A fragment source for this lane: row = row0 + l15, K offset half*2
  const float* aRow = A + (size_t)(row0 + l15) * K + half * 2;

  v8f acc[8] = {};
  for (int k0 = 0; k0 < K; k0 += 4) {
    v2f a = *reinterpret_cast<const v2f*>(aRow + k0);       // A[M=l15][k0+2h .. +1]
    const float* wr0 = &lds[(k0 + half * 2) * COUT];
    const float* wr1 = wr0 + COUT;
#pragma unroll
    for (int ct = 0; ct < 8; ++ct) {
      int col = ct * 16 + l15;
      v2f b;
      b.x = wr0[col];                                       // B[k0+2h][N=l15]
      b.y = wr1[col];                                       // B[k0+2h+1][N=l15]
      // 8-arg form: (neg_a, A, neg_b, B, c_mod, C, reuse_a, reuse_b)
      acc[ct] = __builtin_amdgcn_wmma_f32_16x16x4_f32(
          false, a, false, b, (short)0, acc[ct], false, false);
    }
  }

  // D layout: VGPR r -> row = row0 + r + half*8, col = ct*16 + l15
#pragma unroll
  for (int ct = 0; ct < 8; ++ct) {
    int col = ct * 16 + l15;
    float bv = bias[col];
#pragma unroll
    for (int r = 0; r < 8; ++r) {
      float v = acc[ct][r] + bv;
      if (do_relu) v = fmaxf(v, 0.0f);
      out[(size_t)(row0 + half * 8 + r) * COUT + col] = v;
    }
  }
}

// ---------------------------------------------------------------- driver
extern "C" void kernel_launch(void* const* d_in, const int* in_sizes, int n_in,
                              void* d_out, int out_size, void* d_ws, size_t ws_size,
                              hipStream_t stream) {
  const float* x   = (const float*)d_in[0];
  const int*   ei  = (const int*)d_in[1];
  const float* W1a = (const float*)d_in[2];
  const float* b1a = (const float*)d_in[3];
  const float* W1b = (const float*)d_in[4];
  const float* b1b = (const float*)d_in[5];
  const float* W2a = (const float*)d_in[6];
  const float* b2a = (const float*)d_in[7];
  const float* W2b = (const float*)d_in[8];
  const float* b2b = (const float*)d_in[9];
  float* out = (float*)d_out;

  const int N = in_sizes[0] / 64;   // 50000 (exact multiple of 16)
  const int E = in_sizes[1] / 2;    // 800000
  const int* src = ei;
  const int* dst = ei + E;

  float* R0 = (float*)d_ws;                 // agg1 / agg2
  float* R1 = R0 + (size_t)N * 128;         // t1  / t2
  float* R2 = R1 + (size_t)N * 128;         // h

  const int strips  = (N + 15) / 16;
  const int gblocks = (strips + 7) / 8;

  // ---- layer 1: z1 = x + scatter(x); h = relu(relu(z1 W1a + b1a) W1b + b1b) ----
  {
    int n4 = (N * 64) / 4;
    copy_f4<<<(n4 + 255) / 256, 256, 0, stream>>>(R0, x, n4);
  }
  {
    int total = E * (64 / 4);
    scatter_add<<<(total + 255) / 256, 256, 0, stream>>>(R0, x, src, dst, total, 4, 64);
  }
  gemm_bias_act<<<gblocks, 256,  64 * COUT * sizeof(float), stream>>>(R0, W1a, b1a, R1, N,  64, 1);
  gemm_bias_act<<<gblocks, 256, 128 * COUT * sizeof(float), stream>>>(R1, W1b, b1b, R2, N, 128, 1);

  // ---- layer 2: z2 = h + scatter(h); out = relu(z2 W2a + b2a) W2b + b2b ----
  {
    int n4 = (N * 128) / 4;
    copy_f4<<<(n4 + 255) / 256, 256, 0, stream>>>(R0, R2, n4);
  }
  {
    int total = E * (128 / 4);
    scatter_add<<<(total + 255) / 256, 256, 0, stream>>>(R0, R2, src, dst, total, 5, 128);
  }
  gemm_bias_act<<<gblocks, 256, 128 * COUT * sizeof(float), stream>>>(R0, W2a, b2a, R1, N, 128, 1);
  gemm_bias_act<<<gblocks, 256, 128 * COUT * sizeof(float), stream>>>(R1, W2b, b2b, out, N, 128, 0);
}